// YOLOLayer_20796231647680
// MI455X (gfx1250) — compile-verified
//
#include <hip/hip_runtime.h>
#include <hip/hip_bf16.h>
#include <math.h>

#define A_ANCH 3
#define NCLS   80
#define BN     16
#define GN     32
#define NGT    (BN*GN)       // 512
#define EPSF   1e-7f
#define IGN    0.7f
#define NETSZ  608.0f

typedef float v2f __attribute__((ext_vector_type(2)));
typedef float v8f __attribute__((ext_vector_type(8)));

// Wave-wide (32-lane) sum using V_WMMA_F32_16X16X4_F32.
// A layout (16x4 f32): lanes 0-15 VGPR0 = (M=lane,K=0), VGPR1 = K=1;
// lanes 16-31 VGPR0 = (M=lane-16,K=2), VGPR1 = K=3. With a.y = 0 and B = ones:
// D[m][n] = p_m + p_{m+16} for all n. Per-lane sum of the 8 D VGPRs gives
// lanes<16: sum rows 0-7, lanes>=16: sum rows 8-15; one xor-16 shuffle totals it.
// Requires EXEC all ones: call unconditionally from full 256-thread blocks.
__device__ __forceinline__ float wave_sum32(float v) {
  v2f a; a[0] = v;    a[1] = 0.0f;
  v2f b; b[0] = 1.0f; b[1] = 1.0f;
  v8f c = {};
  c = __builtin_amdgcn_wmma_f32_16x16x4_f32(false, a, false, b, (short)0, c, false, false);
  float r = ((c[0] + c[1]) + (c[2] + c[3])) + ((c[4] + c[5]) + (c[6] + c[7]));
  r += __shfl_xor(r, 16, 32);
  return r;
}

__device__ __forceinline__ float sigmoidf_(float x) { return 1.0f / (1.0f + __expf(-x)); }

// ---- Kernel 1: GT-vs-prior matching + per-GT precompute ----------------------
__global__ void match_kernel(const float* __restrict__ gtb, const float* __restrict__ priors,
                             int* __restrict__ matches, float* __restrict__ gtd) {
  int i = blockIdx.x * blockDim.x + threadIdx.x;
  if (i >= NGT) return;
  float x0 = gtb[i*4+0], y0 = gtb[i*4+1], x1 = gtb[i*4+2], y1 = gtb[i*4+3];
  float ag = (x1 - x0) * (y1 - y0);
  float best = -1e30f; int bj = 0;
  for (int j = 0; j < 9; ++j) {
    float p0 = priors[j*4+0], p1 = priors[j*4+1], p2 = priors[j*4+2], p3 = priors[j*4+3];
    float ap = (p2 - p0) * (p3 - p1);
    float lx = fmaxf(x0, p0), ly = fmaxf(y0, p1);
    float rx = fminf(x1, p2), ry = fminf(y1, p3);
    float w = fmaxf(rx - lx, 0.0f), h = fmaxf(ry - ly, 0.0f);
    float inter = w * h;
    float iou = inter / (ag + ap - inter + 1e-9f);
    if (iou > best) { best = iou; bj = j; }   // first-max semantics like jnp.argmax
  }
  matches[i] = bj;
  float gcx = (x0 + x1) * 0.5f, gcy = (y0 + y1) * 0.5f;
  float gw = x1 - x0, gh = y1 - y0;
  float p0 = priors[bj*4+0], p1 = priors[bj*4+1], p2 = priors[bj*4+2], p3 = priors[bj*4+3];
  float aw = p2 - p0, ah = p3 - p1;
  gtd[i*5+0] = gcx;
  gtd[i*5+1] = gcy;
  gtd[i*5+2] = __logf(gw / (aw + EPSF) + EPSF);
  gtd[i*5+3] = __logf(gh / (ah + EPSF) + EPSF);
  gtd[i*5+4] = 2.0f - gw * gh;
}

// ---- Kernel 2: confidence loss over every predicted box ----------------------
// Templated on S so all /S, %S become constant-divisor mul/shift; anchor index
// comes from blockIdx.z so no runtime division at all. Reads only channels 0..4
// of each 85-channel anchor slice, plane-coalesced within one block.
// grid = (ceil(S*S/256), B, A_ANCH); partials already offset for this fm.
template<int S, int F>
__global__ void conf_kernel(const float* __restrict__ fm, const float* __restrict__ gtb,
                            const float* __restrict__ asz, float* __restrict__ partials) {
  constexpr int HW = S * S;
  const int b = blockIdx.y;
  const int a = blockIdx.z;
  __shared__ float gl[GN][5];
  __shared__ float wsum[8][2];
  int tid = threadIdx.x;
  if (tid < GN) {
    float4 g = ((const float4*)(gtb + (size_t)b * GN * 4))[tid];
    gl[tid][0] = g.x; gl[tid][1] = g.y; gl[tid][2] = g.z; gl[tid][3] = g.w;
    gl[tid][4] = (g.z - g.x) * (g.w - g.y);
  }
  __syncthreads();

  int cell = blockIdx.x * blockDim.x + tid;
  float num = 0.0f, den = 0.0f;
  if (cell < HW) {
    const float* base = fm + ((size_t)b * (A_ANCH*(5+NCLS)) + (size_t)a * (5+NCLS)) * HW + cell;
    float tx = base[0*HW], ty = base[1*HW], tw = base[2*HW], th = base[3*HW], tc = base[4*HW];
    float sx = sigmoidf_(tx), sy = sigmoidf_(ty), sc = sigmoidf_(tc);
    int gx = cell % S, gy = cell / S;              // constant-divisor: mul/shift
    float bx = (sx + (float)gx) * (1.0f / (float)S);
    float by = (sy + (float)gy) * (1.0f / (float)S);
    float bw = asz[(F*3 + a)*2 + 0] * __expf(tw) * (1.0f / NETSZ);
    float bh = asz[(F*3 + a)*2 + 1] * __expf(th) * (1.0f / NETSZ);
    float px0 = bx - bw*0.5f, py0 = by - bh*0.5f;
    float px1 = bx + bw*0.5f, py1 = by + bh*0.5f;
    float pa = (px1 - px0) * (py1 - py0);
    float mx = 0.0f;
    #pragma unroll 8
    for (int g = 0; g < GN; ++g) {
      float lx = fmaxf(px0, gl[g][0]), ly = fmaxf(py0, gl[g][1]);
      float rx = fminf(px1, gl[g][2]), ry = fminf(py1, gl[g][3]);
      float w = fmaxf(rx - lx, 0.0f), h = fmaxf(ry - ly, 0.0f);
      float inter = w * h;
      float iou = inter / (pa + gl[g][4] - inter + 1e-9f);
      mx = fmaxf(mx, iou);
    }
    if (mx < IGN) {
      den = 1.0f;
      num = -__logf(fminf(fmaxf(1.0f - sc, EPSF), 1.0f));
    }
  }
  // full-wave WMMA reduction (all 256 threads participate; EXEC all ones)
  float wn = wave_sum32(num);
  float wd = wave_sum32(den);
  int wave = tid >> 5;
  if ((tid & 31) == 0) { wsum[wave][0] = wn; wsum[wave][1] = wd; }
  __syncthreads();
  if (tid == 0) {
    float n = 0.0f, d = 0.0f;
    int nw = blockDim.x >> 5;
    for (int q = 0; q < nw; ++q) { n += wsum[q][0]; d += wsum[q][1]; }
    int slot = (blockIdx.z * gridDim.y + blockIdx.y) * gridDim.x + blockIdx.x;
    partials[slot*2 + 0] = n;
    partials[slot*2 + 1] = d;
  }
}

// ---- Kernel 3: reg + cls loss over 3*512 matched rows ------------------------
__global__ void regcls_kernel(const float* __restrict__ fm0, const float* __restrict__ fm1,
                              const float* __restrict__ fm2, const int* __restrict__ gcls,
                              const int* __restrict__ matches, const float* __restrict__ gtd,
                              float* __restrict__ partials) {
  int r = blockIdx.x * blockDim.x + threadIdx.x;
  float reg = 0.0f, cls = 0.0f, wv = 0.0f;
  if (r < 3 * NGT) {
    int f = r / NGT;
    int i = r - f * NGT;
    int b = i >> 5;
    int m = matches[i];
    int lo = f * 3;
    if (m >= lo && m < lo + 3) {
      int S = (f == 0) ? 19 : ((f == 1) ? 38 : 76);
      const float* fm = (f == 0) ? fm0 : ((f == 1) ? fm1 : fm2);
      int HW = S * S;
      float gcx = gtd[i*5+0], gcy = gtd[i*5+1];
      float twt = gtd[i*5+2], tht = gtd[i*5+3], s = gtd[i*5+4];
      float fx = floorf(gcx * (float)S);
      float fy = floorf(gcy * (float)S);
      int ix = min(max((int)fx, 0), S - 1);
      int jy = min(max((int)fy, 0), S - 1);
      int cell = min(max(ix * S + jy, 0), HW - 1);   // reference's ix*W + jy quirk
      int loc = m - lo;
      const float* base = fm + ((size_t)b * (A_ANCH*(5+NCLS)) + (size_t)loc * (5+NCLS)) * HW + cell;
      float dx = sigmoidf_(base[0*HW]);
      float dy = sigmoidf_(base[1*HW]);
      float dw = base[2*HW];
      float dh = base[3*HW];
      float tx = gcx * (float)S - fx;
      float ty = gcy * (float)S - fy;
      float d0 = dx - tx, d1 = dy - ty, d2 = dw - twt, d3 = dh - tht;
      reg = s * s * (d0*d0 + d1*d1 + d2*d2 + d3*d3) * 0.5f;
      int cl = gcls[i];
      float p = fminf(fmaxf(sigmoidf_(base[(size_t)(5 + cl) * HW]), EPSF), 1.0f);
      cls = -__logf(p);
      wv = 1.0f;
    }
  }
  float a = wave_sum32(reg);
  float c2 = wave_sum32(cls);
  float c3 = wave_sum32(wv);
  __shared__ float sm[8][3];
  int tid = threadIdx.x, w = tid >> 5;
  if ((tid & 31) == 0) { sm[w][0] = a; sm[w][1] = c2; sm[w][2] = c3; }
  __syncthreads();
  if (tid == 0) {
    float x = 0, y = 0, z = 0;
    for (int q = 0; q < (int)(blockDim.x >> 5); ++q) { x += sm[q][0]; y += sm[q][1]; z += sm[q][2]; }
    partials[blockIdx.x*3 + 0] = x;
    partials[blockIdx.x*3 + 1] = y;
    partials[blockIdx.x*3 + 2] = z;
  }
}

// ---- Kernel 4: fixed-order final fold (deterministic) ------------------------
__global__ void final_kernel(const float* __restrict__ confp, int nconf,
                             const float* __restrict__ regp, int nreg,
                             float* __restrict__ out) {
  __shared__ float sm[8][2];
  int tid = threadIdx.x;
  float n = 0.0f, d = 0.0f;
  for (int s = tid; s < nconf; s += blockDim.x) { n += confp[s*2 + 0]; d += confp[s*2 + 1]; }
  float wn = wave_sum32(n);
  float wd = wave_sum32(d);
  int w = tid >> 5;
  if ((tid & 31) == 0) { sm[w][0] = wn; sm[w][1] = wd; }
  __syncthreads();
  if (tid == 0) {
    float cn = 0, cd = 0;
    for (int q = 0; q < (int)(blockDim.x >> 5); ++q) { cn += sm[q][0]; cd += sm[q][1]; }
    float rn = 0, cln = 0, ws = 0;
    for (int q = 0; q < nreg; ++q) { rn += regp[q*3+0]; cln += regp[q*3+1]; ws += regp[q*3+2]; }
    out[0] = rn / ws + cln / ws + cn / cd;
  }
}

extern "C" void kernel_launch(void* const* d_in, const int* in_sizes, int n_in,
                              void* d_out, int out_size, void* d_ws, size_t ws_size,
                              hipStream_t stream) {
  const float* fm0    = (const float*)d_in[0];
  const float* fm1    = (const float*)d_in[1];
  const float* fm2    = (const float*)d_in[2];
  const float* gtb    = (const float*)d_in[3];
  const int*   gtc    = (const int*)d_in[4];
  const float* priors = (const float*)d_in[5];
  const float* asz    = (const float*)d_in[6];
  float* out = (float*)d_out;

  // workspace layout (everything is fully written before being read; no zeroing)
  int*   matches = (int*)d_ws;                 // 512 ints
  float* gtd     = (float*)d_ws + NGT;         // 512*5 floats
  float* confp   = gtd + NGT * 5;              // 2 * totConfSlots floats

  constexpr int S0 = 19, S1 = 38, S2 = 76;
  const int cb0 = (S0*S0 + 255) / 256;         // 2
  const int cb1 = (S1*S1 + 255) / 256;         // 6
  const int cb2 = (S2*S2 + 255) / 256;         // 23
  const int slots0 = cb0 * BN * A_ANCH;        // 96
  const int slots1 = cb1 * BN * A_ANCH;        // 288
  const int slots2 = cb2 * BN * A_ANCH;        // 1104
  const int tot = slots0 + slots1 + slots2;    // 1488
  float* regp = confp + 2 * tot;               // 6*3 floats

  match_kernel<<<2, 256, 0, stream>>>(gtb, priors, matches, gtd);
  conf_kernel<S0, 0><<<dim3(cb0, BN, A_ANCH), 256, 0, stream>>>(fm0, gtb, asz, confp);
  conf_kernel<S1, 1><<<dim3(cb1, BN, A_ANCH), 256, 0, stream>>>(fm1, gtb, asz, confp + 2*slots0);
  conf_kernel<S2, 2><<<dim3(cb2, BN, A_ANCH), 256, 0, stream>>>(fm2, gtb, asz, confp + 2*(slots0 + slots1));
  regcls_kernel<<<(3*NGT + 255)/256, 256, 0, stream>>>(fm0, fm1, fm2, gtc, matches, gtd, regp);
  final_kernel<<<1, 256, 0, stream>>>(confp, tot, regp, (3*NGT + 255)/256, out);
}